// SageModelfull_1709396984375
// MI455X (gfx1250) — compile-verified
//
#include <hip/hip_runtime.h>
#include <math.h>

#define NNODES 100000
#define NEDGES 1600000
#define DFEAT  128
#define NCLASS 47
#define LN_EPS 1e-5f
#define LDSS   132   // padded LDS row stride (floats): conflict-free b64 A-fragment reads

typedef __attribute__((ext_vector_type(2))) float v2f;
typedef __attribute__((ext_vector_type(8))) float v8f;

// low 32 bits of a generic pointer into LDS == LDS byte offset (ISA §10.2)
__device__ __forceinline__ unsigned lds_addr_u32(const void* p) {
  return (unsigned)(unsigned long long)p;
}

// ---------------------------------------------------------------- utilities
__global__ void zero_f32(float* __restrict__ p, int n) {
  int i = blockIdx.x * blockDim.x + threadIdx.x;
  int stride = gridDim.x * blockDim.x;
  for (; i < n; i += stride) p[i] = 0.0f;
}

__global__ void degree_kernel(const int* __restrict__ dst, float* __restrict__ deg, int e) {
  int i = blockIdx.x * blockDim.x + threadIdx.x;
  if (i < e) atomicAdd(&deg[dst[i]], 1.0f);
}

__global__ void invdeg_kernel(float* __restrict__ deg, int n) {
  int i = blockIdx.x * blockDim.x + threadIdx.x;
  if (i < n) deg[i] = 1.0f / fmaxf(deg[i], 1.0f);
}

// --------------------------------------------------- edge gather/scatter-add
// One wave32 per edge: lane l moves float4 chunk l of the 128-float row.
__global__ void scatter_kernel(const float* __restrict__ x,
                               const int* __restrict__ src,
                               const int* __restrict__ dst,
                               float* __restrict__ agg, int e) {
  long long idx = (long long)blockIdx.x * blockDim.x + threadIdx.x;
  int edge = (int)(idx >> 5);
  int lane = (int)(idx & 31);
  if (edge >= e) return;

  int s = src[edge];
  int d = dst[edge];

  // Prefetch a future edge's source row (global_prefetch_b8)
  int pe = edge + 64;
  if (pe < e) {
    const float* pf = x + (long long)src[pe] * DFEAT + lane * 4;
    __builtin_prefetch(pf, 0, 0);
  }

  const float4 v = *(const float4*)(x + (long long)s * DFEAT + lane * 4);
  float* a = agg + (long long)d * DFEAT + lane * 4;
  atomicAdd(a + 0, v.x);
  atomicAdd(a + 1, v.y);
  atomicAdd(a + 2, v.z);
  atomicAdd(a + 3, v.w);
}

// ------------------------------------------------------- fused SAGE + LN/ReLU
// out[N, NCOLS] = X @ Ws + diag(inv_deg) * (AGG @ Wn) + bias (+ LN + ReLU)
// Block = 256 threads (8 waves): block owns 16 rows, wave w owns cols
// [16w, 16w+16).  X/AGG tiles staged raw into LDS via async DMA
// (global_load_async_to_lds_b128, ASYNCcnt); inv_deg applied to the
// neighbor accumulator post-GEMM.  V_WMMA_F32_16X16X4_F32.
template <int NCOLS, bool DO_LN>
__global__ __launch_bounds__(256)
void sage_gemm(const float* __restrict__ X,
               const float* __restrict__ AGG,
               const float* __restrict__ INVDEG,
               const float* __restrict__ Ws,
               const float* __restrict__ Wn,
               const float* __restrict__ bias,
               const float* __restrict__ gamma,
               const float* __restrict__ beta,
               float* __restrict__ out) {
  __shared__ float ldsX[16 * LDSS];
  __shared__ float ldsG[16 * LDSS];
  __shared__ float tile[16 * LDSS];
  __shared__ float inv_s[16];
  __shared__ float mu_s[16];
  __shared__ float rs_s[16];

  const int tid  = threadIdx.x;
  const int wave = tid >> 5;
  const int lane = tid & 31;
  const int half = lane >> 4;   // 0: K = k+0/k+1 ; 1: K = k+2/k+3
  const int l16  = lane & 15;
  const int row0 = blockIdx.x * 16;
  const int col0 = wave * 16;

  // ---- async DMA stage: raw X and AGG tiles -> LDS (no VGPR round-trip) ----
  // 16 rows x 32 float4 chunks = 512 chunks per matrix; 256 threads x 2.
  {
    const unsigned lXb = lds_addr_u32(ldsX);
    const unsigned lGb = lds_addr_u32(ldsG);
#pragma unroll
    for (int it = 0; it < 2; ++it) {
      int idx = tid + it * 256;
      int r  = idx >> 5;          // 0..15
      int c4 = idx & 31;          // float4 chunk
      unsigned voff = (unsigned)((row0 + r) * 512 + c4 * 16);       // bytes
      unsigned lX = lXb + (unsigned)((r * LDSS + c4 * 4) * 4);      // 16B aligned
      unsigned lG = lGb + (unsigned)((r * LDSS + c4 * 4) * 4);
      asm volatile("global_load_async_to_lds_b128 %0, %1, %2 offset:0"
                   :: "v"(lX), "v"(voff), "s"(X) : "memory");
      asm volatile("global_load_async_to_lds_b128 %0, %1, %2 offset:0"
                   :: "v"(lG), "v"(voff), "s"(AGG) : "memory");
    }
  }
  if (tid < 16) inv_s[tid] = INVDEG[row0 + tid];
  asm volatile("s_wait_asynccnt 0" ::: "memory");
  __syncthreads();

  if (col0 >= NCOLS) return;   // whole-wave early out (no barriers after in !DO_LN)

  const int   bcol  = col0 + l16;
  const int   bcolc = (bcol < NCOLS) ? bcol : (NCOLS - 1);  // clamp, branchless
  const float bmask = (bcol < NCOLS) ? 1.0f : 0.0f;

  const float* ldsa = ldsX + l16 * LDSS;
  const float* ldsg = ldsG + l16 * LDSS;

  v8f acc_s = {0.f, 0.f, 0.f, 0.f, 0.f, 0.f, 0.f, 0.f};
  v8f acc_n = {0.f, 0.f, 0.f, 0.f, 0.f, 0.f, 0.f, 0.f};

#pragma unroll 8
  for (int k = 0; k < 128; k += 4) {
    const int ka = k + 2 * half;

    v2f a_s = *(const v2f*)(ldsa + ka);   // ds_load_b64, bank-conflict-free
    v2f a_n = *(const v2f*)(ldsg + ka);

    v2f b_s, b_n;
    b_s.x = Ws[ka * NCOLS + bcolc] * bmask;
    b_s.y = Ws[(ka + 1) * NCOLS + bcolc] * bmask;
    b_n.x = Wn[ka * NCOLS + bcolc] * bmask;
    b_n.y = Wn[(ka + 1) * NCOLS + bcolc] * bmask;

    acc_s = __builtin_amdgcn_wmma_f32_16x16x4_f32(false, a_s, false, b_s,
                                                  (short)0, acc_s, false, false);
    acc_n = __builtin_amdgcn_wmma_f32_16x16x4_f32(false, a_n, false, b_n,
                                                  (short)0, acc_n, false, false);
  }

  // combine: self + inv_deg(row) * neighbor + bias
  // acc element i is (row = row0 + i + 8*half, col = bcol)
  const float bv = bias[bcolc] * bmask;
  float res[8];
#pragma unroll
  for (int i = 0; i < 8; ++i)
    res[i] = acc_s[i] + inv_s[i + 8 * half] * acc_n[i] + bv;

  if (!DO_LN) {
    if (bcol < NCOLS) {
#pragma unroll
      for (int i = 0; i < 8; ++i) {
        int r = row0 + i + 8 * half;
        out[(long long)r * NCOLS + bcol] = res[i];
      }
    }
    return;
  }

  // ---- LayerNorm + ReLU over the 16x128 stripe via LDS ----
#pragma unroll
  for (int i = 0; i < 8; ++i)
    tile[(i + 8 * half) * LDSS + bcol] = res[i];
  __syncthreads();

  if (tid < 16) {
    float s = 0.f, ss = 0.f;
    const float* rowp = tile + tid * LDSS;
    for (int c = 0; c < 128; ++c) {
      float v = rowp[c];
      s += v; ss += v * v;
    }
    float mu  = s * (1.0f / 128.0f);
    float var = ss * (1.0f / 128.0f) - mu * mu;
    mu_s[tid] = mu;
    rs_s[tid] = rsqrtf(var + LN_EPS);
  }
  __syncthreads();

  for (int e = tid; e < 16 * 128; e += 256) {
    int r = e >> 7, c = e & 127;
    float v = (tile[r * LDSS + c] - mu_s[r]) * rs_s[r] * gamma[c] + beta[c];
    out[(long long)(row0 + r) * 128 + c] = fmaxf(v, 0.0f);
  }
}

// ------------------------------------------------------------------- driver
extern "C" void kernel_launch(void* const* d_in, const int* in_sizes, int n_in,
                              void* d_out, int out_size, void* d_ws, size_t ws_size,
                              hipStream_t stream) {
  (void)in_sizes; (void)n_in; (void)out_size; (void)ws_size;

  const float* h   = (const float*)d_in[0];
  const int*   src = (const int*)d_in[1];
  const int*   dst = (const int*)d_in[2];
  const float* Ws0 = (const float*)d_in[3];
  const float* Wn0 = (const float*)d_in[4];
  const float* b0  = (const float*)d_in[5];
  const float* g0  = (const float*)d_in[6];
  const float* be0 = (const float*)d_in[7];
  const float* Ws1 = (const float*)d_in[8];
  const float* Wn1 = (const float*)d_in[9];
  const float* b1  = (const float*)d_in[10];
  const float* g1  = (const float*)d_in[11];
  const float* be1 = (const float*)d_in[12];
  const float* Ws2 = (const float*)d_in[13];
  const float* Wn2 = (const float*)d_in[14];
  const float* b2  = (const float*)d_in[15];

  float* ws  = (float*)d_ws;
  float* deg = ws;                                   // N (becomes 1/max(deg,1))
  float* agg = deg + NNODES;                         // N*128
  float* x1  = agg + (long long)NNODES * DFEAT;      // N*128
  float* x2  = x1  + (long long)NNODES * DFEAT;      // N*128

  const int T = 256;
  const int FEAT_ELEMS = NNODES * DFEAT;
  const int scatter_blocks = (int)(((long long)NEDGES * 32 + T - 1) / T);
  dim3 gemm_grid(NNODES / 16);   // N is a multiple of 16

  // degrees (shared by all layers)
  zero_f32<<<1024, T, 0, stream>>>(deg, NNODES);
  degree_kernel<<<(NEDGES + T - 1) / T, T, 0, stream>>>(dst, deg, NEDGES);
  invdeg_kernel<<<(NNODES + T - 1) / T, T, 0, stream>>>(deg, NNODES);

  // layer 0: D=128 -> H=128, LN+ReLU
  zero_f32<<<4096, T, 0, stream>>>(agg, FEAT_ELEMS);
  scatter_kernel<<<scatter_blocks, T, 0, stream>>>(h, src, dst, agg, NEDGES);
  sage_gemm<128, true><<<gemm_grid, T, 0, stream>>>(h, agg, deg, Ws0, Wn0, b0,
                                                    g0, be0, x1);

  // layer 1: H=128 -> H=128, LN+ReLU
  zero_f32<<<4096, T, 0, stream>>>(agg, FEAT_ELEMS);
  scatter_kernel<<<scatter_blocks, T, 0, stream>>>(x1, src, dst, agg, NEDGES);
  sage_gemm<128, true><<<gemm_grid, T, 0, stream>>>(x1, agg, deg, Ws1, Wn1, b1,
                                                    g1, be1, x2);

  // layer 2: H=128 -> C=47, plain logits
  zero_f32<<<4096, T, 0, stream>>>(agg, FEAT_ELEMS);
  scatter_kernel<<<scatter_blocks, T, 0, stream>>>(x2, src, dst, agg, NEDGES);
  sage_gemm<NCLASS, false><<<gemm_grid, T, 0, stream>>>(x2, agg, deg, Ws2, Wn2,
                                                        b2, nullptr, nullptr,
                                                        (float*)d_out);
}